// ParallelMLP_58944131170535
// MI455X (gfx1250) — compile-verified
//
#include <hip/hip_runtime.h>
#include <math.h>

typedef __attribute__((ext_vector_type(2))) float v2f;
typedef __attribute__((ext_vector_type(8))) float v8f;

// Problem constants (from reference)
#define NE    8          // experts
#define TOPK  2
#define TT    8192       // tokens = BS*SL
#define HS    1024
#define FFN   4096
#define CAP   2048       // expert capacity
#define NENT  (TT*TOPK)  // 16384 routing entries

// GEMM tiling: block tile 64x128, 8 waves, each wave owns 2x2 16x16 WMMA tiles
#define BM 64
#define BN 128
#define KC 32
#define SA_STRIDE 36     // floats; 144B row stride (16B aligned, even for v2f)
#define SB_STRIDE 132    // floats; 528B row stride (16B aligned)
#define A_ELEMS (BM * SA_STRIDE)
#define B_ELEMS (KC * SB_STRIDE)
#define A_BYTES (A_ELEMS * 4)
#define B_BYTES (B_ELEMS * 4)

__device__ __forceinline__ float gelu_tanh(float v) {
    const float c0 = 0.7978845608028654f;  // sqrt(2/pi)
    const float c1 = 0.044715f;
    return 0.5f * v * (1.0f + tanhf(c0 * (v + c1 * v * v * v)));
}

// 16-byte async copy global -> LDS (ASYNCcnt-tracked, no VGPR staging)
__device__ __forceinline__ void async_cp16(const float* gaddr, unsigned lds_off) {
    asm volatile("global_load_async_to_lds_b128 %0, %1, off"
                 :: "v"(lds_off), "v"(gaddr)
                 : "memory");
}
__device__ __forceinline__ void wait_async0() {
    asm volatile("s_wait_asynccnt 0x0" ::: "memory");
}

// ---------------------------------------------------------------------------
// Fills
// ---------------------------------------------------------------------------
__global__ void fill_neg1_kernel(int* __restrict__ p, int n) {
    int i = blockIdx.x * blockDim.x + threadIdx.x;
    if (i < n) p[i] = -1;
}
__global__ void fill_zero_kernel(float* __restrict__ p, int n) {
    int i = blockIdx.x * blockDim.x + threadIdx.x;
    if (i < n) p[i] = 0.0f;
}

// ---------------------------------------------------------------------------
// Routing: one block per expert. Stable ranks == jnp.argsort(stable) semantics.
// ---------------------------------------------------------------------------
__global__ void route_kernel(const int* __restrict__ te,
                             int* __restrict__ assign,
                             int* __restrict__ row_token) {
    __shared__ int s[256];
    const int e = blockIdx.x;
    int base = 0;
    for (int c = 0; c < NENT; c += 256) {
        const int i = c + threadIdx.x;
        const int flag = (te[i] == e) ? 1 : 0;
        s[threadIdx.x] = flag;
        __syncthreads();
        for (int off = 1; off < 256; off <<= 1) {
            int v = (threadIdx.x >= (unsigned)off) ? s[threadIdx.x - off] : 0;
            __syncthreads();
            s[threadIdx.x] += v;
            __syncthreads();
        }
        const int incl  = s[threadIdx.x];
        const int total = s[255];
        if (flag) {
            const int pos = base + incl - 1;
            if (pos < CAP) {
                assign[i] = e * CAP + pos;
                row_token[e * CAP + pos] = i >> 1;   // token = entry / TOPK
            } else {
                assign[i] = -1;
            }
        }
        base += total;
        __syncthreads();
    }
}

// ---------------------------------------------------------------------------
// GEMM1: h = gelu(gather(x) @ w1), async double-buffered LDS pipeline.
// ---------------------------------------------------------------------------
__global__ void __launch_bounds__(256)
gemm1_gelu_kernel(const float* __restrict__ x,
                  const int* __restrict__ row_token,
                  const float* __restrict__ zbuf,
                  const float* __restrict__ w1,
                  float* __restrict__ hbuf) {
    __shared__ float sA[2 * A_ELEMS];
    __shared__ float sB[2 * B_ELEMS];
    __shared__ const float* s_src[BM];

    const int e       = blockIdx.z;
    const int rowBase = blockIdx.y * BM;
    const int nBase   = blockIdx.x * BN;

    const int tid  = threadIdx.x;
    const int lane = tid & 31;
    const int wave = tid >> 5;
    const int mb   = (wave >> 2) * 32;       // wave M offset: 0/32
    const int nb   = (wave & 3) * 32;        // wave N offset: 0..96
    const int half = lane >> 4;
    const int lm   = lane & 15;

    if (tid < BM) {
        int tok = row_token[e * CAP + rowBase + tid];
        s_src[tid] = (tok >= 0) ? (x + (size_t)tok * HS) : zbuf;
    }
    __syncthreads();

    const unsigned sAoff = (unsigned)(uintptr_t)&sA[0];
    const unsigned sBoff = (unsigned)(uintptr_t)&sB[0];

    // Per-thread fixed async-copy slots (advance by k each chunk)
    const float* gA[2]; unsigned lA[2];
    #pragma unroll
    for (int i = 0; i < 2; ++i) {                 // A: 64 rows x 8 float4
        const int l = tid + i * 256;
        const int r = l >> 3, cc = l & 7;
        gA[i] = s_src[r] + cc * 4;
        lA[i] = sAoff + (unsigned)(r * SA_STRIDE + cc * 4) * 4u;
    }
    const float* wB = w1 + (size_t)e * HS * FFN + nBase;
    const float* gB[4]; unsigned lB[4];
    #pragma unroll
    for (int i = 0; i < 4; ++i) {                 // B: 32 rows x 32 float4
        const int l = tid + i * 256;
        const int kk = l >> 5, cc = l & 31;
        gB[i] = wB + (size_t)kk * FFN + cc * 4;
        lB[i] = sBoff + (unsigned)(kk * SB_STRIDE + cc * 4) * 4u;
    }

    const int NCH = HS / KC;                      // 32 chunks
    // Prologue: issue chunk 0 into buffer 0
    #pragma unroll
    for (int i = 0; i < 2; ++i) async_cp16(gA[i], lA[i]);
    #pragma unroll
    for (int i = 0; i < 4; ++i) async_cp16(gB[i], lB[i]);

    v8f acc00 = {}, acc01 = {}, acc10 = {}, acc11 = {};
    for (int c = 0; c < NCH; ++c) {
        const int buf = c & 1;
        wait_async0();
        __syncthreads();
        if (c + 1 < NCH) {                        // prefetch next chunk
            const int nbuf = (c + 1) & 1;
            const int k0 = (c + 1) * KC;
            #pragma unroll
            for (int i = 0; i < 2; ++i)
                async_cp16(gA[i] + k0, lA[i] + (unsigned)nbuf * A_BYTES);
            #pragma unroll
            for (int i = 0; i < 4; ++i)
                async_cp16(gB[i] + (size_t)k0 * FFN, lB[i] + (unsigned)nbuf * B_BYTES);
        }
        const float* bA = sA + buf * A_ELEMS;
        const float* bB = sB + buf * B_ELEMS;
        #pragma unroll
        for (int kk = 0; kk < KC; kk += 4) {
            v2f a0 = *(const v2f*)&bA[(mb + lm) * SA_STRIDE + kk + 2 * half];
            v2f a1 = *(const v2f*)&bA[(mb + 16 + lm) * SA_STRIDE + kk + 2 * half];
            v2f b0, b1;
            b0.x = bB[(kk + 2 * half) * SB_STRIDE + nb + lm];
            b0.y = bB[(kk + 1 + 2 * half) * SB_STRIDE + nb + lm];
            b1.x = bB[(kk + 2 * half) * SB_STRIDE + nb + 16 + lm];
            b1.y = bB[(kk + 1 + 2 * half) * SB_STRIDE + nb + 16 + lm];
            acc00 = __builtin_amdgcn_wmma_f32_16x16x4_f32(false, a0, false, b0, (short)0, acc00, false, false);
            acc01 = __builtin_amdgcn_wmma_f32_16x16x4_f32(false, a0, false, b1, (short)0, acc01, false, false);
            acc10 = __builtin_amdgcn_wmma_f32_16x16x4_f32(false, a1, false, b0, (short)0, acc10, false, false);
            acc11 = __builtin_amdgcn_wmma_f32_16x16x4_f32(false, a1, false, b1, (short)0, acc11, false, false);
        }
        __syncthreads();
    }

    // Epilogue: GELU + store. C/D layout: VGPR v -> M = v + 8*half, N = lane%16
    float* hp = hbuf + ((size_t)e * CAP + rowBase + mb) * FFN + nBase + nb + lm;
    #pragma unroll
    for (int v = 0; v < 8; ++v) {
        const size_t m0 = (size_t)(v + 8 * half) * FFN;
        const size_t m1 = (size_t)(16 + v + 8 * half) * FFN;
        hp[m0]      = gelu_tanh(acc00[v]);
        hp[m0 + 16] = gelu_tanh(acc01[v]);
        hp[m1]      = gelu_tanh(acc10[v]);
        hp[m1 + 16] = gelu_tanh(acc11[v]);
    }
}

// ---------------------------------------------------------------------------
// GEMM2: obuf = h @ w2, same async pipeline, direct A.
// ---------------------------------------------------------------------------
__global__ void __launch_bounds__(256)
gemm2_kernel(const float* __restrict__ hbuf,
             const float* __restrict__ w2,
             float* __restrict__ obuf) {
    __shared__ float sA[2 * A_ELEMS];
    __shared__ float sB[2 * B_ELEMS];

    const int e       = blockIdx.z;
    const int rowBase = blockIdx.y * BM;
    const int nBase   = blockIdx.x * BN;

    const int tid  = threadIdx.x;
    const int lane = tid & 31;
    const int wave = tid >> 5;
    const int mb   = (wave >> 2) * 32;
    const int nb   = (wave & 3) * 32;
    const int half = lane >> 4;
    const int lm   = lane & 15;

    const unsigned sAoff = (unsigned)(uintptr_t)&sA[0];
    const unsigned sBoff = (unsigned)(uintptr_t)&sB[0];

    const float* aP = hbuf + ((size_t)e * CAP + rowBase) * FFN;
    const float* gA[2]; unsigned lA[2];
    #pragma unroll
    for (int i = 0; i < 2; ++i) {
        const int l = tid + i * 256;
        const int r = l >> 3, cc = l & 7;
        gA[i] = aP + (size_t)r * FFN + cc * 4;
        lA[i] = sAoff + (unsigned)(r * SA_STRIDE + cc * 4) * 4u;
    }
    const float* wB = w2 + (size_t)e * FFN * HS + nBase;
    const float* gB[4]; unsigned lB[4];
    #pragma unroll
    for (int i = 0; i < 4; ++i) {
        const int l = tid + i * 256;
        const int kk = l >> 5, cc = l & 31;
        gB[i] = wB + (size_t)kk * HS + cc * 4;
        lB[i] = sBoff + (unsigned)(kk * SB_STRIDE + cc * 4) * 4u;
    }

    const int NCH = FFN / KC;                     // 128 chunks
    #pragma unroll
    for (int i = 0; i < 2; ++i) async_cp16(gA[i], lA[i]);
    #pragma unroll
    for (int i = 0; i < 4; ++i) async_cp16(gB[i], lB[i]);

    v8f acc00 = {}, acc01 = {}, acc10 = {}, acc11 = {};
    for (int c = 0; c < NCH; ++c) {
        const int buf = c & 1;
        wait_async0();
        __syncthreads();
        if (c + 1 < NCH) {
            const int nbuf = (c + 1) & 1;
            const int k0 = (c + 1) * KC;
            #pragma unroll
            for (int i = 0; i < 2; ++i)
                async_cp16(gA[i] + k0, lA[i] + (unsigned)nbuf * A_BYTES);
            #pragma unroll
            for (int i = 0; i < 4; ++i)
                async_cp16(gB[i] + (size_t)k0 * HS, lB[i] + (unsigned)nbuf * B_BYTES);
        }
        const float* bA = sA + buf * A_ELEMS;
        const float* bB = sB + buf * B_ELEMS;
        #pragma unroll
        for (int kk = 0; kk < KC; kk += 4) {
            v2f a0 = *(const v2f*)&bA[(mb + lm) * SA_STRIDE + kk + 2 * half];
            v2f a1 = *(const v2f*)&bA[(mb + 16 + lm) * SA_STRIDE + kk + 2 * half];
            v2f b0, b1;
            b0.x = bB[(kk + 2 * half) * SB_STRIDE + nb + lm];
            b0.y = bB[(kk + 1 + 2 * half) * SB_STRIDE + nb + lm];
            b1.x = bB[(kk + 2 * half) * SB_STRIDE + nb + 16 + lm];
            b1.y = bB[(kk + 1 + 2 * half) * SB_STRIDE + nb + 16 + lm];
            acc00 = __builtin_amdgcn_wmma_f32_16x16x4_f32(false, a0, false, b0, (short)0, acc00, false, false);
            acc01 = __builtin_amdgcn_wmma_f32_16x16x4_f32(false, a0, false, b1, (short)0, acc01, false, false);
            acc10 = __builtin_amdgcn_wmma_f32_16x16x4_f32(false, a1, false, b0, (short)0, acc10, false, false);
            acc11 = __builtin_amdgcn_wmma_f32_16x16x4_f32(false, a1, false, b1, (short)0, acc11, false, false);
        }
        __syncthreads();
    }

    float* op = obuf + ((size_t)e * CAP + rowBase + mb) * HS + nBase + nb + lm;
    #pragma unroll
    for (int v = 0; v < 8; ++v) {
        const size_t m0 = (size_t)(v + 8 * half) * HS;
        const size_t m1 = (size_t)(16 + v + 8 * half) * HS;
        op[m0]      = acc00[v];
        op[m0 + 16] = acc01[v];
        op[m1]      = acc10[v];
        op[m1 + 16] = acc11[v];
    }
}

// ---------------------------------------------------------------------------
// Combine (gather, no atomics): y[t] = bias + sum_s w[t,s]*obuf[assign[t,s]]
// ---------------------------------------------------------------------------
__global__ void combine_kernel(const float* __restrict__ obuf,
                               const int* __restrict__ assign,
                               const float* __restrict__ ew,
                               const float* __restrict__ bias,
                               float* __restrict__ y) {
    const int t  = blockIdx.x;
    const int a0 = assign[t * TOPK + 0];
    const int a1 = assign[t * TOPK + 1];
    const float w0 = ew[t * TOPK + 0];
    const float w1v = ew[t * TOPK + 1];
    for (int h = threadIdx.x; h < HS; h += blockDim.x) {
        float acc = bias[h];
        if (a0 >= 0) acc += w0 * obuf[(size_t)a0 * HS + h];
        if (a1 >= 0) acc += w1v * obuf[(size_t)a1 * HS + h];
        y[(size_t)t * HS + h] = acc;
    }
}

// ---------------------------------------------------------------------------
extern "C" void kernel_launch(void* const* d_in, const int* in_sizes, int n_in,
                              void* d_out, int out_size, void* d_ws, size_t ws_size,
                              hipStream_t stream) {
    const float* x    = (const float*)d_in[0];   // (BS,SL,HS)
    const float* ew   = (const float*)d_in[1];   // (T, TOPK)
    const float* w1   = (const float*)d_in[2];   // (E, HS, FFN)
    const float* w2   = (const float*)d_in[3];   // (E, FFN, HS)
    const float* bias = (const float*)d_in[4];   // (HS,)
    const int*   te   = (const int*)d_in[5];     // (T, TOPK)
    float* y = (float*)d_out;

    // Workspace layout
    char* p = (char*)d_ws;
    float* hbuf = (float*)p;                 p += (size_t)NE * CAP * FFN * sizeof(float); // 256 MiB
    float* obuf = (float*)p;                 p += (size_t)NE * CAP * HS  * sizeof(float); //  64 MiB
    float* zbuf = (float*)p;                 p += (size_t)HS * sizeof(float);             // zero row
    int* row_token = (int*)p;                p += (size_t)NE * CAP * sizeof(int);
    int* assign    = (int*)p;                p += (size_t)NENT * sizeof(int);
    (void)ws_size; (void)in_sizes; (void)n_in; (void)out_size;

    fill_neg1_kernel<<<(NE * CAP + 255) / 256, 256, 0, stream>>>(row_token, NE * CAP);
    fill_zero_kernel<<<(HS + 255) / 256, 256, 0, stream>>>(zbuf, HS);

    route_kernel<<<NE, 256, 0, stream>>>(te, assign, row_token);

    dim3 g1(FFN / BN, CAP / BM, NE);
    gemm1_gelu_kernel<<<g1, 256, 0, stream>>>(x, row_token, zbuf, w1, hbuf);

    dim3 g2(HS / BN, CAP / BM, NE);
    gemm2_kernel<<<g2, 256, 0, stream>>>(hbuf, w2, obuf);

    combine_kernel<<<TT, 256, 0, stream>>>(obuf, assign, ew, bias, y);
}